// NGNN_14190571946143
// MI455X (gfx1250) — compile-verified
//
#include <hip/hip_runtime.h>
#include <math.h>

typedef __attribute__((ext_vector_type(2))) float v2f;
typedef __attribute__((ext_vector_type(8))) float v8f;

#define N_NODES 50000
#define F_IN    256
#define F_OUT   128
#define LDS_PAD 4   // row stride 260 floats -> conflict-free ds_load_b64 A reads

// Native CDNA5 fp32 scatter-add: non-returning global atomic, device scope
// (coherent across WGPs), tracked with STOREcnt. Guarantees we never fall
// back to a CAS loop regardless of HIP atomic lowering flags.
__device__ __forceinline__ void atomic_add_f32_dev(float* p, float v) {
    asm volatile("global_atomic_add_f32 %0, %1, off scope:SCOPE_DEV"
                 :
                 : "v"(p), "v"(v)
                 : "memory");
}

// ---------------------------------------------------------------------------
// Dense projection + tanh:  out[50000,128] = tanh(A[50000,256] @ W[256,128])
// Block = 256 threads (8 waves). Block owns a 16-row band; wave w owns cols
// [16w, 16w+16). A band staged once in LDS, shared by all 8 waves.
// K-loop: 64 x v_wmma_f32_16x16x4_f32.
// ---------------------------------------------------------------------------
__global__ __launch_bounds__(256)
void ngnn_gemm_tanh(const float* __restrict__ A,
                    const float* __restrict__ W,
                    float* __restrict__ out) {
    __shared__ float As[16][F_IN + LDS_PAD];

    const int tid  = threadIdx.x;
    const int row0 = blockIdx.x * 16;

    // Stage 16x256 fp32 tile: 1024 float4 slots / 256 threads = 4 each.
    const float4* Av = (const float4*)(A + (size_t)row0 * F_IN);
#pragma unroll
    for (int i = 0; i < 4; ++i) {
        int idx = tid + i * 256;          // float4 slot in 16x64 grid
        int r   = idx >> 6;               // 64 float4 per row
        int c   = idx & 63;
        float4 v = Av[r * 64 + c];
        As[r][c * 4 + 0] = v.x;
        As[r][c * 4 + 1] = v.y;
        As[r][c * 4 + 2] = v.z;
        As[r][c * 4 + 3] = v.w;
    }
    __syncthreads();

    const int wave = tid >> 5;
    const int lane = tid & 31;
    const int col0 = wave * 16;
    const int half = lane >> 4;           // 0: K+0,K+1   1: K+2,K+3
    const int m    = lane & 15;           // A: row index / B: col index

    v8f acc = {};
#pragma unroll 8
    for (int k = 0; k < F_IN; k += 4) {
        const int ka = k + 2 * half;
        // A fragment 16x4: lane(0-15)=row m holds K=ka,ka+1 ; 8B-aligned LDS read
        v2f a = *(const v2f*)&As[m][ka];
        // B fragment 4x16: lane holds col m for K=ka,ka+1 ; coalesced global reads
        v2f b;
        b[0] = W[(size_t)ka * F_OUT + col0 + m];
        b[1] = W[(size_t)(ka + 1) * F_OUT + col0 + m];
        acc = __builtin_amdgcn_wmma_f32_16x16x4_f32(
            /*neg_a=*/false, a, /*neg_b=*/false, b,
            /*c_mod=*/(short)0, acc, /*reuse_a=*/false, /*reuse_b=*/false);
    }

    // C/D layout: vgpr r, lanes 0-15 -> M=r ; lanes 16-31 -> M=r+8 ; N = lane&15
#pragma unroll
    for (int r = 0; r < 8; ++r) {
        int M = r + half * 8;
        out[(size_t)(row0 + M) * F_OUT + col0 + m] = tanhf(acc[r]);
    }
}

// ---------------------------------------------------------------------------
// Edge-scatter SpMM hop: out[dst[e]] += w[e] * x[src[e]]   (128 features)
// One wave per edge; lane handles a float4 -> coalesced 512B row gather and
// 4 native fp32 global atomic adds per lane. The whole working set (x, out,
// indices, weights ~70MB) lives in the 192MB L2, so this runs at L2 rate.
// ---------------------------------------------------------------------------
__global__ __launch_bounds__(256)
void ngnn_spmm(const float* __restrict__ x,
               float* __restrict__ out,
               const long long* __restrict__ src,
               const long long* __restrict__ dst,
               const float* __restrict__ ew,
               int E) {
    const int gw   = (int)((blockIdx.x * (unsigned)blockDim.x + threadIdx.x) >> 5);
    const int lane = threadIdx.x & 31;
    if (gw >= E) return;

    const long long s = src[gw];
    const long long d = dst[gw];
    const float     w = ew[gw];

    const float4* xv = (const float4*)(x + (size_t)s * F_OUT);
    float4 v = xv[lane];

    float* o = out + (size_t)d * F_OUT + lane * 4;
    atomic_add_f32_dev(o + 0, v.x * w);
    atomic_add_f32_dev(o + 1, v.y * w);
    atomic_add_f32_dev(o + 2, v.z * w);
    atomic_add_f32_dev(o + 3, v.w * w);
}

// ---------------------------------------------------------------------------
__global__ __launch_bounds__(256)
void ngnn_fill_zero(float4* __restrict__ p, int n4) {
    int i = blockIdx.x * blockDim.x + threadIdx.x;
    if (i < n4) p[i] = make_float4(0.f, 0.f, 0.f, 0.f);
}

__global__ __launch_bounds__(256)
void ngnn_bias_init(float4* __restrict__ out, const float4* __restrict__ bias, int n4) {
    int i = blockIdx.x * blockDim.x + threadIdx.x;
    if (i < n4) out[i] = bias[i & 31];   // 128 floats = 32 float4 per node row
}

// ---------------------------------------------------------------------------
extern "C" void kernel_launch(void* const* d_in, const int* in_sizes, int n_in,
                              void* d_out, int out_size, void* d_ws, size_t ws_size,
                              hipStream_t stream) {
    const float*     features = (const float*)d_in[0];       // [50000,256] f32
    const float*     weight   = (const float*)d_in[1];       // [256,128]  f32
    const float*     bias     = (const float*)d_in[2];       // [128]      f32
    const long long* eidx     = (const long long*)d_in[3];   // [2,E]      i64
    const float*     ew       = (const float*)d_in[4];       // [E]        f32

    const int E = in_sizes[4];
    const long long* srcIdx = eidx;        // row 0
    const long long* dstIdx = eidx + E;    // row 1

    float* buf0 = (float*)d_ws;                              // 25.6 MB
    float* buf1 = buf0 + (size_t)N_NODES * F_OUT;            // 25.6 MB
    float* out  = (float*)d_out;

    const int TOT4       = (N_NODES * F_OUT) / 4;            // 1,600,000 float4
    const int fillBlocks = (TOT4 + 255) / 256;
    const int spmmBlocks = (E + 7) / 8;                      // 8 waves/block

    // 1) H = tanh(F @ W)   -> buf0
    ngnn_gemm_tanh<<<N_NODES / 16, 256, 0, stream>>>(features, weight, buf0);

    // 2) hop 1: buf1 = Adj * buf0
    ngnn_fill_zero<<<fillBlocks, 256, 0, stream>>>((float4*)buf1, TOT4);
    ngnn_spmm<<<spmmBlocks, 256, 0, stream>>>(buf0, buf1, srcIdx, dstIdx, ew, E);

    // 3) hop 2: buf0 = Adj * buf1
    ngnn_fill_zero<<<fillBlocks, 256, 0, stream>>>((float4*)buf0, TOT4);
    ngnn_spmm<<<spmmBlocks, 256, 0, stream>>>(buf1, buf0, srcIdx, dstIdx, ew, E);

    // 4) hop 3 scatters straight into d_out, pre-seeded with the bias
    ngnn_bias_init<<<fillBlocks, 256, 0, stream>>>((float4*)out, (const float4*)bias, TOT4);
    ngnn_spmm<<<spmmBlocks, 256, 0, stream>>>(buf0, out, srcIdx, dstIdx, ew, E);
}